// BVH_11751030522280
// MI455X (gfx1250) — compile-verified
//
#include <hip/hip_runtime.h>
#include <math.h>

typedef float v2f __attribute__((ext_vector_type(2)));
typedef float v8f __attribute__((ext_vector_type(8)));

#define WAVES_PER_BLOCK 8
#define TPB (WAVES_PER_BLOCK * 32)
#define NCHUNKS 8

static __device__ __forceinline__ v8f splat8(float x) {
  v8f r = {x, x, x, x, x, x, x, x};
  return r;
}

// D = A(16x4 f32) * B(4x16 f32) + C(16x16 f32), wave32 WMMA
static __device__ __forceinline__ v8f wmma4(v2f a, v2f b, v8f c) {
  return __builtin_amdgcn_wmma_f32_16x16x4_f32(
      /*neg_a=*/false, a, /*neg_b=*/false, b,
      /*c_mod=*/(short)0, c, /*reuse_a=*/false, /*reuse_b=*/false);
}

// n / where(|d|<1e-30, 1, d)  via fast reciprocal
static __device__ __forceinline__ float rcp_safe(float d) {
  float ds = (__builtin_fabsf(d) < 1e-30f) ? 1.0f : d;
  return __builtin_amdgcn_rcpf(ds);
}

// Stage 1: each block = 16 points x one triangle-chunk. Produces one candidate
// (key,u,v,w,idx) per point per chunk into workspace (or, in direct mode with a
// single chunk, writes the final outputs immediately).
__global__ __launch_bounds__(TPB) void closest_tri_stage1(
    const float* __restrict__ tris,   // [F,3,3]
    const float* __restrict__ pts,    // [Q,3]
    float* __restrict__ out,          // [Q | 3Q | Q | 3Q] (direct mode only)
    float* __restrict__ wsK, float* __restrict__ wsU,
    float* __restrict__ wsV, float* __restrict__ wsW,
    int* __restrict__ wsI,
    int F, int Q, int chunks, int tilesPerChunk, int direct) {
  __shared__ float sK[WAVES_PER_BLOCK * 16];
  __shared__ float sU[WAVES_PER_BLOCK * 16];
  __shared__ float sV[WAVES_PER_BLOCK * 16];
  __shared__ float sW[WAVES_PER_BLOCK * 16];
  __shared__ int   sI[WAVES_PER_BLOCK * 16];

  const int tid  = threadIdx.x;
  const int wave = tid >> 5;
  const int lane = tid & 31;
  const int half = lane >> 4;   // 0: lanes 0-15 (rows 0-7, K=0/1), 1: lanes 16-31 (rows 8-15, K=2/3)
  const int nl   = lane & 15;   // triangle column within tile / point row for A
  const int pBase = blockIdx.x * 16;
  const int chunk = blockIdx.y;

  // ---- A matrix: 16 points x 4 (x,y,z,0). Lanes 0-15 hold K=0,1; lanes 16-31 hold K=2,3.
  int qm = pBase + nl;
  if (qm > Q - 1) qm = Q - 1;
  float px = pts[qm * 3 + 0], py = pts[qm * 3 + 1], pz = pts[qm * 3 + 2];
  v2f A;
  A.x = half ? pz : px;
  A.y = half ? 0.0f : py;

  // running best per accumulator row (key = cp^2 - 2*cp.p ; p^2 added at writeout)
  float bk[8], bu[8], bv[8], bw[8];
  int bi[8];
#pragma unroll
  for (int r = 0; r < 8; ++r) {
    bk[r] = __builtin_inff();
    bu[r] = 0.0f; bv[r] = 0.0f; bw[r] = 0.0f;
    bi[r] = 0x7fffffff;
  }

  const int NT = (F + 15) >> 4;
  int tStart = chunk * tilesPerChunk;
  int tEnd = tStart + tilesPerChunk;
  if (tEnd > NT) tEnd = NT;

  for (int t = tStart + wave; t < tEnd; t += WAVES_PER_BLOCK) {
    int g = t * 16 + nl;
    if (g > F - 1) g = F - 1;  // duplicate last tri on tail; idx tie-break keeps correctness
    const float* T = tris + (size_t)g * 9;

    // prefetch next tile for this wave
    int t2 = t + WAVES_PER_BLOCK;
    if (t2 < tEnd) {
      int g2 = t2 * 16 + nl;
      if (g2 > F - 1) g2 = F - 1;
      __builtin_prefetch(tris + (size_t)g2 * 9, 0, 0);
    }

    float ax = T[0], ay = T[1], az = T[2];
    float bx = T[3], by = T[4], bz = T[5];
    float cx = T[6], cy = T[7], cz = T[8];

    float abx = bx - ax, aby = by - ay, abz = bz - az;
    float acx = cx - ax, acy = cy - ay, acz = cz - az;

    // per-triangle dot constants (broadcast across the 8 point rows)
    float a_ab = ax * abx + ay * aby + az * abz;
    float a_ac = ax * acx + ay * acy + az * acz;
    float b_ab = bx * abx + by * aby + bz * abz;
    float b_ac = bx * acx + by * acy + bz * acz;
    float c_ab = cx * abx + cy * aby + cz * abz;
    float c_ac = cx * acx + cy * acy + cz * acz;
    float aa  = ax * ax + ay * ay + az * az;
    float bb  = bx * bx + by * by + bz * bz;
    float cc  = cx * cx + cy * cy + cz * cz;
    float abd = ax * bx + ay * by + az * bz;
    float acd = ax * cx + ay * cy + az * cz;
    float bcd = bx * cx + by * cy + bz * cz;

    // B matrices 4x16: lane%16 = column N; lanes 0-15 hold K=0(V0),K=1(V1); lanes 16-31 hold K=2,K=3(=0 pad)
    v2f Bab; Bab.x = half ? abz : abx; Bab.y = half ? 0.0f : aby;
    v2f Bac; Bac.x = half ? acz : acx; Bac.y = half ? 0.0f : acy;
    v2f Ba;  Ba.x  = half ? az  : ax;  Ba.y  = half ? 0.0f : ay;
    v2f Bb;  Bb.x  = half ? bz  : bx;  Bb.y  = half ? 0.0f : by;
    v2f Bc;  Bc.x  = half ? cz  : cx;  Bc.y  = half ? 0.0f : cy;

    // d1..d6 directly out of WMMA (constant folded into accumulator), plus p.a, p.b, p.c
    v8f D1 = wmma4(A, Bab, splat8(-a_ab));
    v8f D2 = wmma4(A, Bac, splat8(-a_ac));
    v8f D3 = wmma4(A, Bab, splat8(-b_ab));
    v8f D4 = wmma4(A, Bac, splat8(-b_ac));
    v8f D5 = wmma4(A, Bab, splat8(-c_ab));
    v8f D6 = wmma4(A, Bac, splat8(-c_ac));
    v8f Z  = splat8(0.0f);
    v8f PA = wmma4(A, Ba, Z);
    v8f PB = wmma4(A, Bb, Z);
    v8f PC = wmma4(A, Bc, Z);

#pragma unroll
    for (int r = 0; r < 8; ++r) {
      float d1 = D1[r], d2 = D2[r], d3 = D3[r], d4 = D4[r], d5 = D5[r], d6 = D6[r];
      float pa = PA[r], pb = PB[r], pc = PC[r];

      float va = d3 * d6 - d5 * d4;
      float vb = d5 * d2 - d1 * d6;
      float vc = d1 * d4 - d3 * d2;
      float rden = rcp_safe(va + vb + vc);
      float v_ = vb * rden;
      float w_ = vc * rden;
      float u_ = 1.0f - v_ - w_;

      float d43 = d4 - d3, d56 = d5 - d6;
      bool c6 = (va <= 0.0f) && (d43 >= 0.0f) && (d56 >= 0.0f);
      float t6 = d43 * rcp_safe(d43 + d56);
      u_ = c6 ? 0.0f : u_; v_ = c6 ? (1.0f - t6) : v_; w_ = c6 ? t6 : w_;

      bool c5 = (vb <= 0.0f) && (d2 >= 0.0f) && (d6 <= 0.0f);
      float t5 = d2 * rcp_safe(d2 - d6);
      u_ = c5 ? (1.0f - t5) : u_; v_ = c5 ? 0.0f : v_; w_ = c5 ? t5 : w_;

      bool c4 = (d6 >= 0.0f) && (d5 <= d6);
      u_ = c4 ? 0.0f : u_; v_ = c4 ? 0.0f : v_; w_ = c4 ? 1.0f : w_;

      bool c3 = (vc <= 0.0f) && (d1 >= 0.0f) && (d3 <= 0.0f);
      float t3 = d1 * rcp_safe(d1 - d3);
      u_ = c3 ? (1.0f - t3) : u_; v_ = c3 ? t3 : v_; w_ = c3 ? 0.0f : w_;

      bool c2 = (d3 >= 0.0f) && (d4 <= d3);
      u_ = c2 ? 0.0f : u_; v_ = c2 ? 1.0f : v_; w_ = c2 ? 0.0f : w_;

      bool c1 = (d1 <= 0.0f) && (d2 <= 0.0f);
      u_ = c1 ? 1.0f : u_; v_ = c1 ? 0.0f : v_; w_ = c1 ? 0.0f : w_;

      float cpd = u_ * pa + v_ * pb + w_ * pc;
      float cp2 = u_ * u_ * aa + v_ * v_ * bb + w_ * w_ * cc +
                  2.0f * (u_ * v_ * abd + u_ * w_ * acd + v_ * w_ * bcd);
      float key = cp2 - 2.0f * cpd;

      bool better = (key < bk[r]) || ((key == bk[r]) && (g < bi[r]));
      bk[r] = better ? key : bk[r];
      bu[r] = better ? u_ : bu[r];
      bv[r] = better ? v_ : bv[r];
      bw[r] = better ? w_ : bw[r];
      bi[r] = better ? g : bi[r];
    }
  }

  // ---- intra-wave reduction: min over the 16 triangle-columns per point row
#pragma unroll
  for (int r = 0; r < 8; ++r) {
    float k = bk[r], uu = bu[r], vv = bv[r], ww = bw[r];
    int id = bi[r];
#pragma unroll
    for (int off = 8; off >= 1; off >>= 1) {
      float k2 = __shfl_xor(k, off, 32);
      float u2 = __shfl_xor(uu, off, 32);
      float v2 = __shfl_xor(vv, off, 32);
      float w2 = __shfl_xor(ww, off, 32);
      int   i2 = __shfl_xor(id, off, 32);
      bool take = (k2 < k) || ((k2 == k) && (i2 < id));
      k  = take ? k2 : k;
      uu = take ? u2 : uu;
      vv = take ? v2 : vv;
      ww = take ? w2 : ww;
      id = take ? i2 : id;
    }
    if (nl == 0) {  // lanes 0 and 16 hold results for points r and 8+r
      int m = half * 8 + r;
      int s = wave * 16 + m;
      sK[s] = k; sU[s] = uu; sV[s] = vv; sW[s] = ww; sI[s] = id;
    }
  }
  __syncthreads();

  // ---- cross-wave reduction (16 points per block)
  if (tid < 16) {
    int m = tid;
    float k = sK[m], uu = sU[m], vv = sV[m], ww = sW[m];
    int id = sI[m];
    for (int wv = 1; wv < WAVES_PER_BLOCK; ++wv) {
      int s = wv * 16 + m;
      float k2 = sK[s];
      int   i2 = sI[s];
      bool take = (k2 < k) || ((k2 == k) && (i2 < id));
      if (take) { k = k2; uu = sU[s]; vv = sV[s]; ww = sW[s]; id = i2; }
    }
    int q = pBase + m;
    if (q < Q) {
      if (direct) {
        // single-chunk fallback: produce final outputs here
        float qx = pts[q * 3 + 0], qy = pts[q * 3 + 1], qz = pts[q * 3 + 2];
        float p2 = qx * qx + qy * qy + qz * qz;
        float dist = p2 + k;
        dist = dist > 0.0f ? dist : 0.0f;
        const float* T = tris + (size_t)id * 9;
        out[q] = dist;
        out[Q + q * 3 + 0] = uu * T[0] + vv * T[3] + ww * T[6];
        out[Q + q * 3 + 1] = uu * T[1] + vv * T[4] + ww * T[7];
        out[Q + q * 3 + 2] = uu * T[2] + vv * T[5] + ww * T[8];
        out[4 * Q + q] = (float)id;
        out[5 * Q + q * 3 + 0] = uu;
        out[5 * Q + q * 3 + 1] = vv;
        out[5 * Q + q * 3 + 2] = ww;
      } else {
        int s = q * chunks + chunk;
        wsK[s] = k; wsU[s] = uu; wsV[s] = vv; wsW[s] = ww; wsI[s] = id;
      }
    }
  }
}

// Stage 2: reduce the per-chunk candidates for each point and emit outputs.
__global__ __launch_bounds__(256) void closest_tri_stage2(
    const float* __restrict__ tris, const float* __restrict__ pts,
    float* __restrict__ out,
    const float* __restrict__ wsK, const float* __restrict__ wsU,
    const float* __restrict__ wsV, const float* __restrict__ wsW,
    const int* __restrict__ wsI,
    int Q, int chunks) {
  int q = blockIdx.x * blockDim.x + threadIdx.x;
  if (q >= Q) return;
  int base = q * chunks;
  float k = wsK[base], uu = wsU[base], vv = wsV[base], ww = wsW[base];
  int id = wsI[base];
  for (int c = 1; c < chunks; ++c) {
    float k2 = wsK[base + c];
    int   i2 = wsI[base + c];
    bool take = (k2 < k) || ((k2 == k) && (i2 < id));
    if (take) { k = k2; uu = wsU[base + c]; vv = wsV[base + c]; ww = wsW[base + c]; id = i2; }
  }
  float qx = pts[q * 3 + 0], qy = pts[q * 3 + 1], qz = pts[q * 3 + 2];
  float p2 = qx * qx + qy * qy + qz * qz;
  float dist = p2 + k;
  dist = dist > 0.0f ? dist : 0.0f;
  const float* T = tris + (size_t)id * 9;
  out[q] = dist;                                  // distances
  out[Q + q * 3 + 0] = uu * T[0] + vv * T[3] + ww * T[6];   // closest_points
  out[Q + q * 3 + 1] = uu * T[1] + vv * T[4] + ww * T[7];
  out[Q + q * 3 + 2] = uu * T[2] + vv * T[5] + ww * T[8];
  out[4 * Q + q] = (float)id;                     // closest_faces
  out[5 * Q + q * 3 + 0] = uu;                    // closest_bcs
  out[5 * Q + q * 3 + 1] = vv;
  out[5 * Q + q * 3 + 2] = ww;
}

extern "C" void kernel_launch(void* const* d_in, const int* in_sizes, int n_in,
                              void* d_out, int out_size, void* d_ws, size_t ws_size,
                              hipStream_t stream) {
  (void)n_in; (void)out_size;
  const float* tris = (const float*)d_in[0];  // [1,F,3,3] f32
  const float* pts  = (const float*)d_in[1];  // [1,Q,3]  f32
  float* out = (float*)d_out;
  int F = in_sizes[0] / 9;
  int Q = in_sizes[1] / 3;
  int NT = (F + 15) >> 4;

  // Split triangles across chunks for occupancy; fall back to direct single-pass
  // if the workspace can't hold the per-chunk candidates.
  int chunks = NCHUNKS;
  size_t need = (size_t)Q * (size_t)chunks * 5u * sizeof(float);
  int direct = (d_ws == nullptr || ws_size < need) ? 1 : 0;
  if (direct) chunks = 1;
  int tilesPerChunk = (NT + chunks - 1) / chunks;

  float* ws = (float*)d_ws;
  size_t qc = (size_t)Q * (size_t)chunks;
  float* wsK = ws;
  float* wsU = ws + qc;
  float* wsV = ws + 2 * qc;
  float* wsW = ws + 3 * qc;
  int*   wsI = (int*)(ws + 4 * qc);

  dim3 grid1((Q + 15) / 16, chunks);
  dim3 block1(TPB);
  hipLaunchKernelGGL(closest_tri_stage1, grid1, block1, 0, stream,
                     tris, pts, out, wsK, wsU, wsV, wsW, wsI,
                     F, Q, chunks, tilesPerChunk, direct);
  if (!direct) {
    dim3 grid2((Q + 255) / 256);
    dim3 block2(256);
    hipLaunchKernelGGL(closest_tri_stage2, grid2, block2, 0, stream,
                       tris, pts, out, wsK, wsU, wsV, wsW, wsI, Q, chunks);
  }
}